// EventCameraSim_28673201668117
// MI455X (gfx1250) — compile-verified
//
#include <hip/hip_runtime.h>
#include <math.h>

// EventCameraSim: elementwise expansion [H,W,C] -> [H,W,C,K] time-of-event map.
// Memory-bound: 531 MB NT store stream, ~24 us floor at 23.3 TB/s.

typedef __attribute__((ext_vector_type(4))) float v4f;

#define HH   720
#define WW   1280
#define CCH  3
#define KK   48
#define NPIX (HH * WW * CCH)     // 2,764,800 = 10800 * 256 exactly
#define CTHV 0.15f
#define EPSV 1e-3f

__global__ __launch_bounds__(256) void event_sim_kernel(
    const float* __restrict__ x,
    const float* __restrict__ img0,
    const float* __restrict__ tptr,
    float* __restrict__ out)
{
    const int p = blockIdx.x * 256 + threadIdx.x;
    if (p >= NPIX) return;   // exact grid, guard is free

    // Inputs are streamed once -> non-temporal loads (TH=NT keeps L2 for reuse elsewhere).
    const float I0 = __builtin_nontemporal_load(img0 + p);
    const float X  = __builtin_nontemporal_load(x + p);
    const float dt = tptr[0];                 // delta_t = time - T0, T0 == 0

    // Per-pixel scalars (match reference arithmetic order for bit-faithful masking).
    const float It  = logf(I0 + EPSV);        // safe_log(initial_image)
    const float xl  = logf(X + EPSV);         // safe_log(x)
    const float dI  = xl - It;
    const float nev = floorf(fabsf(dI / CTHV));
    const float sgn = (dI > 0.0f) ? 1.0f : ((dI < 0.0f) ? -1.0f : 0.0f);
    const float pol = (nev > 0.0f) ? sgn : 0.0f;
    const float slope      = dI / dt;
    const float slope_safe = (fabsf(slope) > 0.0f) ? slope : 1.0f;

    const float base = pol * CTHV;            // img_interp = base * k
    const float r    = base / slope_safe;     // t_all      = r * k (one divide per pixel)
    const bool  pos  = pol > 0.0f;
    const bool  neg  = pol < 0.0f;
    const float NANF = __uint_as_float(0x7FC00000u);

    // 48 contiguous f32 per pixel = 12 x b128 stores; wave32 writes a contiguous 6 KB span.
    v4f* o = reinterpret_cast<v4f*>(out) + (size_t)p * (KK / 4);

    #pragma unroll
    for (int i = 0; i < KK / 4; ++i) {
        v4f v;
        #pragma unroll
        for (int j = 0; j < 4; ++j) {
            const float kf   = (float)(i * 4 + j + 1);  // karange = 1..K
            const float img  = base * kf;               // pol * k * C
            const float s    = img + It;                // exactly as reference: img + It <> xl
            const bool  cond = (pos && (s < xl)) || (neg && (s > xl));
            v[j] = cond ? (r * kf) : NANF;
        }
        // 531 MB >> 192 MB L2: non-temporal so the output stream doesn't thrash L2.
        __builtin_nontemporal_store(v, o + i);
    }
}

extern "C" void kernel_launch(void* const* d_in, const int* in_sizes, int n_in,
                              void* d_out, int out_size, void* d_ws, size_t ws_size,
                              hipStream_t stream) {
    const float* x    = (const float*)d_in[0];   // x:             [H,W,C] f32
    const float* img0 = (const float*)d_in[1];   // initial_image: [H,W,C] f32
    const float* t    = (const float*)d_in[2];   // time: scalar f32 (1 element)
    float* out        = (float*)d_out;           // [H,W,C,K] f32

    dim3 block(256);
    dim3 grid((NPIX + 255) / 256);               // 10800 blocks, no tail
    event_sim_kernel<<<grid, block, 0, stream>>>(x, img0, t, out);
}